// RNNDecoder_7206955123378
// MI455X (gfx1250) — compile-verified
//
#include <hip/hip_runtime.h>
#include <stdint.h>

// CDNA5 / gfx1250, wave32. Full-fp32 WMMA (V_WMMA_F32_16X16X4_F32) +
// async global->LDS copies (GLOBAL_LOAD_ASYNC_TO_LDS_B128) with double buffer.
typedef __attribute__((ext_vector_type(2))) float v2f;
typedef __attribute__((ext_vector_type(8))) float v8f;

#define B_  128
#define P_  196
#define F_  2048
#define E_  512
#define H_  512
#define V_  10000
#define T_  32
#define TS_ 31          // T-1
#define MT_ (TS_ * B_)  // 3968 rows for the time-batched GEMMs

// ---------------- mean pooling over P ----------------
__global__ void mean_pool_kernel(const float* __restrict__ feat,
                                 float* __restrict__ out) {
  int idx = blockIdx.x * blockDim.x + threadIdx.x;   // b*F + f
  if (idx >= B_ * F_) return;
  int b = idx / F_, f = idx - b * F_;
  const float* p = feat + (size_t)b * P_ * F_ + f;
  float s = 0.f;
  for (int i = 0; i < P_; ++i) s += p[(size_t)i * F_];
  out[idx] = s * (1.0f / (float)P_);
}

// ---------------- embedding gather: row r = t*128 + b ----------------
__global__ void gather_kernel(const float* __restrict__ embW,
                              const int* __restrict__ cap,
                              float* __restrict__ out) {
  int r = blockIdx.x;                // 0..MT_-1
  int t = r >> 7, b = r & 127;
  int tok = cap[b * T_ + t];
  const float4* src = (const float4*)(embW + (size_t)tok * E_);
  float4* dst = (float4*)(out + (size_t)r * E_);
  dst[threadIdx.x] = src[threadIdx.x];   // blockDim.x == E_/4 == 128
}

// Async 16B global -> LDS copy (VGPR-bypassing, tracked by ASYNCcnt).
__device__ __forceinline__ void async_cp16(unsigned lds_off, const float* g) {
  unsigned long long ga = (unsigned long long)g;
  asm volatile("global_load_async_to_lds_b128 %0, %1, off"
               :: "v"(lds_off), "v"(ga) : "memory");
}
__device__ __forceinline__ void wait_async0() {
  asm volatile("s_wait_asynccnt 0x0" ::: "memory");
}

// ---------------- generic WMMA-f32 GEMM ----------------
// C[M,N] = A[M,K] * W[N,K]^T (+bias1[N]) (+bias2[N]) (+addend[M,N])
// btv!=0: output index = (row%128)*sb + (row/128)*st + col   (b,t,v layout)
// Block: 256 threads (8 wave32), tile 128(M) x 64(N), K chunked by 32 through
// double-buffered LDS filled with async global->LDS b128 copies; chunk i+1 is
// fetched while the 32 WMMAs of chunk i run. N-edge rows are clamped to a
// valid source row (finite garbage, never stored; epilogue is guarded).
__global__ __launch_bounds__(256) void gemm_wmma(
    const float* __restrict__ A, const float* __restrict__ W,
    const float* __restrict__ bias1, const float* __restrict__ bias2,
    const float* __restrict__ addend, float* __restrict__ C,
    int M, int N, int K, int btv, long sb, long st) {
  __shared__ float As[2][128][36];  // stride 36: conflict-free b64 frag reads,
  __shared__ float Bs[2][64][36];   // 16B-aligned async stores

  const int tid  = threadIdx.x;
  const int lane = tid & 31;
  const int wave = tid >> 5;
  const int half = lane >> 4;      // 0: lanes 0-15, 1: lanes 16-31
  const int sub  = lane & 15;      // M index (A frag) / N index (B frag)
  const int kb   = half << 1;      // K sub-offset 0 or 2 (ISA 16x4 f32 layout)
  const int mblk = blockIdx.y * 128;
  const int nblk = blockIdx.x * 64;
  const int m0w  = wave << 4;

  // Staging geometry: A = 128x32 floats (2 thr/row, 4x b128 each),
  //                   B = 64x32 floats (4 thr/row, 2x b128 each).
  const int arow = tid >> 1;
  const int ac0  = (tid & 1) << 4;
  const int brow = tid >> 2;
  const int bc0  = (tid & 3) << 3;
  const int bn   = nblk + brow;
  const int bnc  = (bn < N) ? bn : (N - 1);            // clamp: stay in bounds

  const float* aSrc = A + (size_t)(mblk + arow) * K + ac0;
  const float* bSrc = W + (size_t)bnc * K + bc0;

  auto issue = [&](int kc, int buf) {
#pragma unroll
    for (int q = 0; q < 4; ++q)
      async_cp16((unsigned)(uintptr_t)&As[buf][arow][ac0 + 4 * q],
                 aSrc + kc + 4 * q);
#pragma unroll
    for (int q = 0; q < 2; ++q)
      async_cp16((unsigned)(uintptr_t)&Bs[buf][brow][bc0 + 4 * q],
                 bSrc + kc + 4 * q);
  };

  v8f acc0 = {}, acc1 = {}, acc2 = {}, acc3 = {};

  issue(0, 0);                               // prologue: fetch chunk 0
  const int nk = K >> 5;                     // K/32 chunks
  for (int i = 0; i < nk; ++i) {
    wait_async0();                           // my writes of chunk i complete
    __syncthreads();                         // everyone's complete; buf[i&1] ready
    if (i + 1 < nk) issue((i + 1) << 5, (i + 1) & 1);   // overlap next fetch
    const int buf = i & 1;
#pragma unroll
    for (int kp = 0; kp < 32; kp += 4) {
      // A fragment 16x4 f32: v0={K=kb}, v1={K=kb+1} per ISA layout
      v2f a  = *(const v2f*)&As[buf][m0w + sub][kp + kb];
      v2f b0 = *(const v2f*)&Bs[buf][ 0 + sub][kp + kb];
      v2f b1 = *(const v2f*)&Bs[buf][16 + sub][kp + kb];
      v2f b2 = *(const v2f*)&Bs[buf][32 + sub][kp + kb];
      v2f b3 = *(const v2f*)&Bs[buf][48 + sub][kp + kb];
      acc0 = __builtin_amdgcn_wmma_f32_16x16x4_f32(false, a, false, b0,
                                                   (short)0, acc0, false, false);
      acc1 = __builtin_amdgcn_wmma_f32_16x16x4_f32(false, a, false, b1,
                                                   (short)0, acc1, false, false);
      acc2 = __builtin_amdgcn_wmma_f32_16x16x4_f32(false, a, false, b2,
                                                   (short)0, acc2, false, false);
      acc3 = __builtin_amdgcn_wmma_f32_16x16x4_f32(false, a, false, b3,
                                                   (short)0, acc3, false, false);
    }
  }

  // Store: D VGPR v -> row mblk + m0w + v + 8*half, col = n0 + sub
  v8f accs[4] = {acc0, acc1, acc2, acc3};
#pragma unroll
  for (int j = 0; j < 4; ++j) {
    int n0 = nblk + (j << 4);
    if (n0 >= N) continue;                 // edge guard only in epilogue
    int col = n0 + sub;
    float bsum = 0.f;
    if (bias1) bsum += bias1[col];
    if (bias2) bsum += bias2[col];
    int rbase = mblk + m0w + (half << 3);
#pragma unroll
    for (int v = 0; v < 8; ++v) {
      int row = rbase + v;
      float val = accs[j][v] + bsum;
      if (addend) val += addend[(size_t)row * N + col];
      size_t off = btv ? ((size_t)(row & 127) * (size_t)sb +
                          (size_t)(row >> 7) * (size_t)st + col)
                       : ((size_t)row * N + col);
      C[off] = val;
    }
  }
}

// ---------------- LSTM elementwise (one time step) ----------------
__device__ __forceinline__ float sigf(float x) { return 1.0f / (1.0f + __expf(-x)); }

__global__ void lstm_elem_kernel(const float* __restrict__ gates,
                                 float* __restrict__ h, float* __restrict__ c,
                                 float* __restrict__ hs_t) {
  int idx = blockIdx.x * blockDim.x + threadIdx.x;  // b*H + j
  if (idx >= B_ * H_) return;
  int b = idx >> 9, j = idx & (H_ - 1);
  const float* g = gates + ((size_t)b << 11);       // b * 4H
  float ig = sigf(g[j]);
  float fg = sigf(g[j + H_]);
  float gg = tanhf(g[j + 2 * H_]);
  float og = sigf(g[j + 3 * H_]);
  float cn = fg * c[idx] + ig * gg;
  float hn = og * tanhf(cn);
  c[idx] = cn;
  h[idx] = hn;
  hs_t[idx] = hn;
}

extern "C" void kernel_launch(void* const* d_in, const int* in_sizes, int n_in,
                              void* d_out, int out_size, void* d_ws, size_t ws_size,
                              hipStream_t stream) {
  const float* features = (const float*)d_in[0];
  const int*   captions = (const int*)d_in[1];
  const float* emb_W = (const float*)d_in[2];
  const float* W_ih  = (const float*)d_in[3];
  const float* W_hh  = (const float*)d_in[4];
  const float* b_ih  = (const float*)d_in[5];
  const float* b_hh  = (const float*)d_in[6];
  const float* Wh    = (const float*)d_in[7];
  const float* bh    = (const float*)d_in[8];
  const float* Wc    = (const float*)d_in[9];
  const float* bc    = (const float*)d_in[10];
  const float* fc_W  = (const float*)d_in[11];
  const float* fc_b  = (const float*)d_in[12];
  float* out = (float*)d_out;

  float* ws = (float*)d_ws;
  float* mean_feat = ws;  ws += (size_t)B_ * F_;        // 1 MB
  float* embeds    = ws;  ws += (size_t)MT_ * E_;       // 8.1 MB
  float* xg        = ws;  ws += (size_t)MT_ * 4 * H_;   // 32.5 MB
  float* gates     = ws;  ws += (size_t)B_ * 4 * H_;    // 1 MB
  float* hbuf      = ws;  ws += (size_t)B_ * H_;
  float* cbuf      = ws;  ws += (size_t)B_ * H_;
  float* hs        = ws;  ws += (size_t)MT_ * H_;       // 8.1 MB

  // 1) mean pooled features
  mean_pool_kernel<<<(B_ * F_ + 255) / 256, 256, 0, stream>>>(features, mean_feat);
  // 2) embedding gather for t = 0..T-2
  gather_kernel<<<MT_, E_ / 4, 0, stream>>>(emb_W, captions, embeds);
  // 3) h0 / c0 init projections (M=128, N=512, K=2048)
  gemm_wmma<<<dim3(H_ / 64, 1), 256, 0, stream>>>(
      mean_feat, Wh, bh, nullptr, nullptr, hbuf, B_, H_, F_, 0, 0, 0);
  gemm_wmma<<<dim3(H_ / 64, 1), 256, 0, stream>>>(
      mean_feat, Wc, bc, nullptr, nullptr, cbuf, B_, H_, F_, 0, 0, 0);
  // 4) x-side gate preactivations for all steps (M=3968, N=2048, K=512)
  gemm_wmma<<<dim3(4 * H_ / 64, MT_ / 128), 256, 0, stream>>>(
      embeds, W_ih, b_ih, b_hh, nullptr, xg, MT_, 4 * H_, E_, 0, 0, 0);
  // 5) recurrent scan: gates = xg[t] + h @ W_hh^T, then elementwise
  for (int t = 0; t < TS_; ++t) {
    gemm_wmma<<<dim3(4 * H_ / 64, 1), 256, 0, stream>>>(
        hbuf, W_hh, nullptr, nullptr, xg + (size_t)t * B_ * 4 * H_,
        gates, B_, 4 * H_, H_, 0, 0, 0);
    lstm_elem_kernel<<<(B_ * H_ + 255) / 256, 256, 0, stream>>>(
        gates, hbuf, cbuf, hs + (size_t)t * B_ * H_);
  }
  // 6) vocab projection with [b,t,v] output remap (M=3968, N=10000, K=512)
  gemm_wmma<<<dim3((V_ + 63) / 64, MT_ / 128), 256, 0, stream>>>(
      hs, fc_W, fc_b, nullptr, nullptr, out, MT_, V_, H_,
      1, (long)TS_ * V_, (long)V_);
}